// PoolNU_79499844649386
// MI455X (gfx1250) — compile-verified
//
#include <hip/hip_runtime.h>
#include <hip/hip_bf16.h>

// PoolNU: out[p, l] = max_{k<9} x[p, nb[k, l]]
//   x  : (B*C = 1024 planes) x LIN=65536  f32   (256 MB)
//   nb : K=9 x LOUT=16384                 int   (indices into LIN)
//   out: 1024 x 16384                     f32   (64 MB)
//
// MI455X plan: one 1024-thread block per plane. Async-DMA the whole 256 KB
// plane into LDS (fits in the WGP's 320 KB), then gather-max out of LDS.
// HBM sees only streaming traffic (~320 MB total -> ~14 us at 23.3 TB/s).

#define LIN   65536
#define LOUT  16384
#define KNB   9
#define TPB   1024

#if defined(__AMDGCN__) && __has_builtin(__builtin_amdgcn_global_load_async_to_lds_b128)
#define HAVE_ASYNC_LDS 1
#else
#define HAVE_ASYNC_LDS 0
#endif

#if HAVE_ASYNC_LDS
// Exact parameter type per hipcc diagnostic: GCC vector of 4 ints,
// global (addrspace(1)) source, LDS (addrspace(3)) destination.
typedef int v4i __attribute__((vector_size(16)));
typedef __attribute__((address_space(1))) v4i* g_v4i_p;
typedef __attribute__((address_space(3))) v4i* l_v4i_p;
#endif

__global__ __launch_bounds__(TPB) void PoolNU_gather_max_kernel(
    const float* __restrict__ x,
    const int*   __restrict__ nb,
    float*       __restrict__ out)
{
    __shared__ float plane[LIN];   // 256 KB of the WGP's 320 KB LDS

    const int    t       = (int)threadIdx.x;
    const size_t planeId = (size_t)blockIdx.x;

    const float4* __restrict__ xp4 = (const float4*)(x + planeId * (size_t)LIN);
    float*        __restrict__ op  = out + planeId * (size_t)LOUT;

    // ---- Stage the full plane into LDS: 16384 float4 / 1024 threads = 16 each,
    //      lane-contiguous 16B chunks -> perfectly coalesced streaming loads.
    {
        float4* p4 = (float4*)plane;
#if HAVE_ASYNC_LDS
#pragma unroll
        for (int i = 0; i < (LIN / 4) / TPB; ++i) {
            const int e4 = i * TPB + t;
            __builtin_amdgcn_global_load_async_to_lds_b128(
                (g_v4i_p)(xp4 + e4),
                (l_v4i_p)(p4 + e4),
                /*imm offset*/0, /*cpol*/0);
        }
#if __has_builtin(__builtin_amdgcn_s_wait_asynccnt)
        __builtin_amdgcn_s_wait_asynccnt(0);
#else
        asm volatile("s_wait_asynccnt 0" ::: "memory");
#endif
#else
#pragma unroll
        for (int i = 0; i < (LIN / 4) / TPB; ++i) {
            const int e4 = i * TPB + t;
            p4[e4] = xp4[e4];
        }
#endif
    }
    __syncthreads();

    // ---- Gather-max out of LDS. nb is K-major: nb[k*LOUT + l], so each of the
    //      9 index loads is coalesced across lanes (and L2-resident after the
    //      first plane). Output stores are coalesced + non-temporal (write-once).
#pragma unroll 2
    for (int l = t; l < LOUT; l += TPB) {
        int   idx0 = nb[0 * LOUT + l];
        float m    = plane[idx0];
#pragma unroll
        for (int k = 1; k < KNB; ++k) {
            int idx = nb[k * LOUT + l];
            m = fmaxf(m, plane[idx]);
        }
        __builtin_nontemporal_store(m, op + l);
    }
}

extern "C" void kernel_launch(void* const* d_in, const int* in_sizes, int n_in,
                              void* d_out, int out_size, void* d_ws, size_t ws_size,
                              hipStream_t stream)
{
    const float* x  = (const float*)d_in[0];
    const int*   nb = (const int*)d_in[1];
    float*       out = (float*)d_out;

    const int planes = in_sizes[0] / LIN;   // 8 * 128 = 1024

    PoolNU_gather_max_kernel<<<dim3(planes), dim3(TPB), 0, stream>>>(x, nb, out);
}